// Covariance_Bayesian_Proxy_Anchor_12163347383263
// MI455X (gfx1250) — compile-verified
//
#include <hip/hip_runtime.h>
#include <hip/hip_bf16.h>

#define BB 384
#define CC 512
#define DD 256

typedef __attribute__((ext_vector_type(2))) float v2f;
typedef __attribute__((ext_vector_type(8))) float v8f;
typedef unsigned int __attribute__((ext_vector_type(4))) u32x4;
typedef int __attribute__((ext_vector_type(4))) i32x4;
typedef int __attribute__((ext_vector_type(8))) i32x8;

// LDS byte offset of a __shared__ object (AS3 pointer value == LDS offset).
__device__ __forceinline__ unsigned lds_off(const void* p) {
  return (unsigned)(size_t)(__attribute__((address_space(3))) const void*)p;
}

__device__ __forceinline__ float warp_red_sum(float v) {
  v += __shfl_xor(v, 1, 32);
  v += __shfl_xor(v, 2, 32);
  v += __shfl_xor(v, 4, 32);
  v += __shfl_xor(v, 8, 32);
  v += __shfl_xor(v, 16, 32);
  return v;
}

// Result valid on thread 0. All threads must call. sred must hold >= blockDim/32 floats.
__device__ __forceinline__ float block_red_sum(float v, float* sred) {
  int lane = threadIdx.x & 31, w = threadIdx.x >> 5;
  v = warp_red_sum(v);
  if (lane == 0) sred[w] = v;
  __syncthreads();
  float r = 0.f;
  if (threadIdx.x < (blockDim.x >> 5)) r = sred[threadIdx.x];
  if (w == 0) r = warp_red_sum(r);
  __syncthreads();
  return r;
}

// ---------------------------------------------------------------------------
// K0: inverse L2 norms of X rows and proxy rows (matches 1/sqrt(sum+1e-12))
// ---------------------------------------------------------------------------
__global__ void __launch_bounds__(DD) k_norms(const float* __restrict__ X,
                                              const float* __restrict__ P,
                                              float* __restrict__ xinvn,
                                              float* __restrict__ pinvn) {
  __shared__ float sred[8];
  int wg = blockIdx.x;
  const float* row = (wg < BB) ? (X + (size_t)wg * DD) : (P + (size_t)(wg - BB) * DD);
  float v = row[threadIdx.x];
  float s = block_red_sum(v * v, sred);
  if (threadIdx.x == 0) {
    float inv = rsqrtf(s + 1e-12f);
    if (wg < BB) xinvn[wg] = inv; else pinvn[wg - BB] = inv;
  }
}

// ---------------------------------------------------------------------------
// K1: per-class counts, means, member index lists
// ---------------------------------------------------------------------------
__global__ void __launch_bounds__(DD) k_classstats(const float* __restrict__ X,
                                                   const int* __restrict__ T,
                                                   float* __restrict__ counts,
                                                   float* __restrict__ meanbuf,
                                                   int* __restrict__ members) {
  __shared__ int scnt;
  int c = blockIdx.x, t = threadIdx.x;
  float s = 0.f;
  int cnt = 0;
  for (int b = 0; b < BB; ++b) {
    if (T[b] == c) {
      s += X[(size_t)b * DD + t];
      if (t == 0) members[c * BB + cnt] = b;
      ++cnt;
    }
  }
  if (t == 0) { scnt = cnt; counts[c] = (float)cnt; }
  __syncthreads();
  int n = scnt;
  meanbuf[(size_t)c * DD + t] = s / (float)(n > 1 ? n : 1);
}

// ---------------------------------------------------------------------------
// K2: build integrated covariance (cov + diag(clamp(gu)) + 1e-6 I) in workspace
// ---------------------------------------------------------------------------
__global__ void __launch_bounds__(DD) k_setup_inv(const float* __restrict__ cov,
                                                  const float* __restrict__ gu,
                                                  float* __restrict__ covinv) {
  int c = blockIdx.x;
  float dadd = fminf(fmaxf(gu[0], 1e-6f), 1.0f) + 1e-6f;
  size_t base = (size_t)c * DD * DD;
  for (int idx = threadIdx.x; idx < DD * DD; idx += DD) {
    float v = cov[base + idx];
    int d = idx >> 8, e = idx & 255;
    if (d == e) v += dadd;
    covinv[base + idx] = v;
  }
}

// ---------------------------------------------------------------------------
// K3: in-place Gauss-Jordan inversion, one workgroup per class. All 512
// matrices (134 MB) are L2-resident (192 MB L2), so the iterative sweeps run
// at L2 bandwidth. Thread j owns column j.
// ---------------------------------------------------------------------------
__global__ void __launch_bounds__(DD) k_gj(float* __restrict__ covinv) {
  int c = blockIdx.x, j = threadIdx.x;
  float* A = covinv + (size_t)c * DD * DD;
  for (int k = 0; k < DD; ++k) {
    float piv = A[k * DD + k];
    float aj  = A[k * DD + j];
    float ip  = 1.0f / piv;
    float rowv = (j == k) ? ip : aj * ip;
    __threadfence(); __syncthreads();
    A[k * DD + j] = rowv;              // scaled pivot row (diag := 1/piv)
    __threadfence(); __syncthreads();
    if (j != k) {
      for (int i = 0; i < DD; ++i) {
        if (i == k) continue;
        float f = A[i * DD + k];       // original column-k values (untouched)
        A[i * DD + j] -= f * rowv;
      }
    }
    __threadfence(); __syncthreads();
    if (j != k) A[j * DD + k] = -A[j * DD + k] * ip;   // fix column k (row j)
    __threadfence(); __syncthreads();
  }
}

// ---------------------------------------------------------------------------
// TDM issue helper: load a 2D tile (tile_dim0 x tile_dim1 f32, row stride
// d0_stride elements) from global into LDS at lds_addr (packed row-major).
// D# layout per CDNA5 ISA §8.3/8.4. Toolchain arity picked via the TDM
// header that only ships with the 6-arg toolchain.
// ---------------------------------------------------------------------------
__device__ __forceinline__ void tdm_load_tile_f32(unsigned long long gaddr,
                                                  unsigned lds_addr,
                                                  unsigned tensor_d0,
                                                  unsigned tensor_d1,
                                                  unsigned tile_d0,
                                                  unsigned tile_d1,
                                                  unsigned d0_stride) {
  u32x4 g0 = {0u, 0u, 0u, 0u};
  g0[0] = 1u;                                   // count=1, user descriptor
  g0[1] = lds_addr;                             // LDS byte address
  g0[2] = (unsigned)gaddr;                      // global addr [31:0]
  g0[3] = ((unsigned)(gaddr >> 32) & 0x01FFFFFFu) | (2u << 30);  // addr[56:32]|type=2
  i32x8 g1 = {0, 0, 0, 0, 0, 0, 0, 0};
  g1[0] = (int)(2u << 16);                      // data_size=2 (4 bytes), mask=0
  g1[1] = (int)(tensor_d0 << 16);               // tensor_dim0[15:0]
  g1[2] = (int)((tensor_d0 >> 16) | (tensor_d1 << 16));  // d0[31:16] | d1[15:0]
  g1[3] = (int)((tensor_d1 >> 16) | (tile_d0 << 16));    // d1[31:16] | tile_dim0
  g1[4] = (int)(tile_d1 & 0xFFFFu);             // tile_dim1 (tile_dim2=0)
  g1[5] = (int)d0_stride;                       // tensor_dim0_stride[31:0]
  i32x4 gz = {0, 0, 0, 0};
#if __has_include(<hip/amd_detail/amd_gfx1250_TDM.h>)
  i32x8 gz8 = {0, 0, 0, 0, 0, 0, 0, 0};
  __builtin_amdgcn_tensor_load_to_lds(g0, g1, gz, gz, gz8, 0);
#else
  __builtin_amdgcn_tensor_load_to_lds(g0, g1, gz, gz, 0);
#endif
}

// ---------------------------------------------------------------------------
// K4: Mahalanobis quad via f32 WMMA + fused cosine + proxy-anchor sums.
// One workgroup per class (8 waves). Each wave owns 3 batch tiles of 16 rows.
// cov_inv N-panels (256x16 f32) are pulled L2->LDS by the Tensor Data Mover
// (one descriptor per panel, issued by wave 0, TENSORcnt-tracked); the proxy
// row is staged with GLOBAL_LOAD_ASYNC_TO_LDS_B128 (ASYNCcnt-tracked).
// B-operand reads panel[k*16+n]: half-wave bank sets {16k+n} / {16k+32+n}
// mod 64 are disjoint -> no padding needed.
// ---------------------------------------------------------------------------
__global__ void __launch_bounds__(256)
k_mahal(const float* __restrict__ X, const int* __restrict__ T,
        const float* __restrict__ P, const float* __restrict__ gu,
        const float* __restrict__ covinv, const float* __restrict__ xinvn,
        const float* __restrict__ pinvn,
        float* __restrict__ Psum, float* __restrict__ Nsum) {
  __shared__ float sprox[DD];
  __shared__ float spP[DD * 16];     // B panel, row-major [k][n]
  __shared__ float sP, sN;

  int c = blockIdx.x, t = threadIdx.x;
  int wave = t >> 5, lane = t & 31;
  int halfsel = lane >> 4;           // 0: lanes 0-15, 1: lanes 16-31
  int l15 = lane & 15;
  int kb = halfsel * 2;              // K sub-offset for A/B operand layout

  if (t == 0) { sP = 0.f; sN = 0.f; }
  // async-stage raw proxy row (1 KB) straight into LDS, 16B per lane
  unsigned prox_base = lds_off(&sprox[0]);
  if (t < 64) {
    unsigned long long g = (unsigned long long)(P + (size_t)c * DD + t * 4);
    unsigned l = prox_base + (unsigned)(t * 16);
    asm volatile("global_load_async_to_lds_b128 %0, %1, off"
                 :: "v"(l), "v"(g) : "memory");
    asm volatile("s_wait_asynccnt 0x0" ::: "memory");
  }
  float gu0 = gu[0];
  float pin = pinvn[c];
  size_t cbase = (size_t)c * DD * DD;
  unsigned panel_base = lds_off(&spP[0]);
  __syncthreads();

  float qacc[3] = {0.f, 0.f, 0.f};

  for (int ntile = 0; ntile < 16; ++ntile) {
    int n0 = ntile * 16;
    // TDM: one descriptor moves the whole 256x16 panel into LDS
    if (wave == 0) {
      unsigned long long g =
          (unsigned long long)(covinv + cbase + (size_t)n0);
      tdm_load_tile_f32(g, panel_base, DD, DD, 16, DD, DD);
      __builtin_amdgcn_s_wait_tensorcnt((unsigned short)0);
    }
    __syncthreads();

    for (int ti = 0; ti < 3; ++ti) {
      int rowbase = (wave + ti * 8) * 16;
      int arow = rowbase + l15;
      const float* xr = X + (size_t)arow * DD;
      const float* bp = spP + l15;   // column n = l15
      __builtin_prefetch(xr, 0, 0);
      v8f y = {0.f, 0.f, 0.f, 0.f, 0.f, 0.f, 0.f, 0.f};
#pragma unroll 8
      for (int k4 = 0; k4 < 64; ++k4) {
        int k = k4 * 4 + kb;
        v2f a, b;
        a.x = xr[k]     - sprox[k];
        a.y = xr[k + 1] - sprox[k + 1];
        b.x = bp[k * 16];
        b.y = bp[k * 16 + 16];
        y = __builtin_amdgcn_wmma_f32_16x16x4_f32(false, a, false, b,
                                                  (short)0, y, false, false);
      }
      // quad contribution: diag of Y @ Diff^T for this N tile
      int col = n0 + l15;
#pragma unroll
      for (int r = 0; r < 8; ++r) {
        int M = rowbase + (halfsel ? (8 + r) : r);
        float dv = X[(size_t)M * DD + col] - sprox[col];
        float p = y[r] * dv;
        p += __shfl_xor(p, 1, 32);
        p += __shfl_xor(p, 2, 32);
        p += __shfl_xor(p, 4, 32);
        p += __shfl_xor(p, 8, 32);             // per-16-lane-half row sums
        float phi = __shfl(p, l15 + 16, 32);   // bring high-half (rows 8+r)
        if (lane == r)     qacc[ti] += p;      // row r   -> lane r
        if (lane == 8 + r) qacc[ti] += phi;    // row 8+r -> lane 8+r
      }
    }
    __syncthreads();
  }

  // finalize: cosine + proxy-anchor exp sums (lane l<16 owns row rowbase+l)
  float accP = 0.f, accN = 0.f;
  for (int ti = 0; ti < 3; ++ti) {
    int rowbase = (wave + ti * 8) * 16;
    int row = rowbase + l15;
    const float* xr = X + (size_t)row * DD;
    float dp = 0.f;
    int ks = halfsel * 128;
#pragma unroll 8
    for (int k = 0; k < 128; ++k) dp += xr[ks + k] * sprox[ks + k];
    dp += __shfl_xor(dp, 16, 32);
    if (lane < 16) {
      float q = fmaxf(qacc[ti], 0.f);
      float mah = __expf(-sqrtf(q) * gu0);
      float cs  = dp * xinvn[row] * pin;
      float s   = 0.6f * mah + 0.4f * cs;
      if (T[row] == c) accP += __expf(-32.0f * (s - 0.1f));
      else             accN += __expf( 32.0f * (s + 0.1f));
    }
  }
  accP = warp_red_sum(accP);
  accN = warp_red_sum(accN);
  if (lane == 0) { atomicAdd(&sP, accP); atomicAdd(&sN, accN); }
  __syncthreads();
  if (t == 0) { Psum[c] = sP; Nsum[c] = sN; }
}

// ---------------------------------------------------------------------------
// K5: streaming pass over class_covariances: cov_reg partials + per-class
// empirical-covariance MSE (only classes with count>1; members staged in LDS).
// Pure bandwidth: 134 MB coalesced reads.
// ---------------------------------------------------------------------------
__global__ void __launch_bounds__(256)
k_cons(const float* __restrict__ X, const float* __restrict__ cov,
       const float* __restrict__ counts, const float* __restrict__ meanbuf,
       const int* __restrict__ members,
       float* __restrict__ covreg, float* __restrict__ cons) {
  __shared__ float sm[16 * DD];
  __shared__ float smu[DD];
  __shared__ float sred[8];
  int c = blockIdx.x, t = threadIdx.x;
  int cnt = (int)counts[c];
  smu[t] = meanbuf[(size_t)c * DD + t];
  __syncthreads();
  int nl = cnt < 16 ? cnt : 16;
  for (int i = 0; i < nl; ++i) {
    int row = members[c * BB + i];
    sm[i * DD + t] = X[(size_t)row * DD + t] - smu[t];
  }
  __syncthreads();
  bool doCons = (cnt > 1);
  float inv = 1.0f / (float)((cnt - 1) > 1 ? (cnt - 1) : 1);
  float aabs = 0.f, amse = 0.f;
  size_t base = (size_t)c * DD * DD;
  for (int idx = t; idx < DD * DD; idx += 256) {
    float cv = cov[base + idx];
    aabs += fabsf(cv);
    if (doCons) {
      int d = idx >> 8, e = idx & 255;   // d uniform per iteration, e == t
      float s = 0.f;
      for (int i = 0; i < nl; ++i) s += sm[i * DD + d] * sm[i * DD + e];
      for (int i = 16; i < cnt; ++i) {   // rare overflow path, X is L2-resident
        int row = members[c * BB + i];
        s += (X[(size_t)row * DD + d] - smu[d]) * (X[(size_t)row * DD + e] - smu[e]);
      }
      float df = s * inv - cv;
      amse += df * df;
    }
  }
  float ra = block_red_sum(aabs, sred);
  float rm = block_red_sum(amse, sred);
  if (t == 0) {
    covreg[c] = ra;
    cons[c] = doCons ? (rm / (float)(DD * DD)) : 0.f;
  }
}

// ---------------------------------------------------------------------------
// K6: final scalar assembly
// ---------------------------------------------------------------------------
__global__ void __launch_bounds__(256)
k_final(const float* __restrict__ Xunc, const float* __restrict__ counts,
        const float* __restrict__ Psum, const float* __restrict__ Nsum,
        const float* __restrict__ covreg, const float* __restrict__ cons,
        const float* __restrict__ gu, float* __restrict__ out) {
  __shared__ float sred[8];
  int t = threadIdx.x;
  float pos = 0.f, neg = 0.f, nv = 0.f, cr = 0.f, cs = 0.f, xu = 0.f;
  for (int c = t; c < CC; c += 256) {
    pos += log1pf(Psum[c]);          // log1p(0)=0 for empty classes
    neg += log1pf(Nsum[c]);
    nv  += (counts[c] != 0.f) ? 1.f : 0.f;
    cr  += covreg[c];
    cs  += cons[c];
  }
  for (int i = t; i < BB * DD; i += 256) xu += Xunc[i];
  pos = block_red_sum(pos, sred);
  neg = block_red_sum(neg, sred);
  nv  = block_red_sum(nv,  sred);
  cr  = block_red_sum(cr,  sred);
  cs  = block_red_sum(cs,  sred);
  xu  = block_red_sum(xu,  sred);
  if (t == 0) {
    float pu = fminf(fmaxf(gu[0], 1e-6f), 1.0f);
    float cov_reg  = cr / (float)((size_t)CC * DD * DD);
    float unc_reg  = xu / (float)(BB * DD) + pu;
    float mainloss = pos / fmaxf(nv, 1.f) + neg / (float)CC;
    out[0] = mainloss + 0.1f * (cov_reg + unc_reg + cs);
  }
}

extern "C" void kernel_launch(void* const* d_in, const int* in_sizes, int n_in,
                              void* d_out, int out_size, void* d_ws, size_t ws_size,
                              hipStream_t stream) {
  const float* X    = (const float*)d_in[0];
  const float* Xunc = (const float*)d_in[1];
  const int*   T    = (const int*)d_in[2];
  const float* Pxy  = (const float*)d_in[3];
  const float* Cov  = (const float*)d_in[4];
  const float* Gu   = (const float*)d_in[5];
  float* out = (float*)d_out;
  (void)in_sizes; (void)n_in; (void)out_size; (void)ws_size;

  float* ws = (float*)d_ws;
  size_t off = 0;
  float* covinv  = ws + off; off += (size_t)CC * DD * DD;  // 134 MB, L2-resident
  float* xinvn   = ws + off; off += BB;
  float* pinvn   = ws + off; off += CC;
  float* counts  = ws + off; off += CC;
  float* meanbuf = ws + off; off += (size_t)CC * DD;
  float* PsumB   = ws + off; off += CC;
  float* NsumB   = ws + off; off += CC;
  float* covregB = ws + off; off += CC;
  float* consB   = ws + off; off += CC;
  int*   members = (int*)(ws + off);

  k_norms<<<BB + CC, DD, 0, stream>>>(X, Pxy, xinvn, pinvn);
  k_classstats<<<CC, DD, 0, stream>>>(X, T, counts, meanbuf, members);
  k_setup_inv<<<CC, DD, 0, stream>>>(Cov, Gu, covinv);
  k_gj<<<CC, DD, 0, stream>>>(covinv);
  k_mahal<<<CC, 256, 0, stream>>>(X, T, Pxy, Gu, covinv, xinvn, pinvn, PsumB, NsumB);
  k_cons<<<CC, 256, 0, stream>>>(X, Cov, counts, meanbuf, members, covregB, consB);
  k_final<<<1, 256, 0, stream>>>(Xunc, counts, PsumB, NsumB, covregB, consB, Gu, out);
}